// GATCritic_66486093742484
// MI455X (gfx1250) — compile-verified
//
#include <hip/hip_runtime.h>
#include <hip/hip_bf16.h>

typedef float v2f __attribute__((ext_vector_type(2)));
typedef float v8f __attribute__((ext_vector_type(8)));

#define NEG_SLOPE 0.2f

// ---------------------------------------------------------------------------
// Utility: fill a float buffer with a constant
// ---------------------------------------------------------------------------
__global__ void fill_f32(float* __restrict__ p, long long n, float v) {
    long long t = (long long)blockIdx.x * blockDim.x + threadIdx.x;
    if (t < n) p[t] = v;
}

// ---------------------------------------------------------------------------
// Layer-1 GEMM: xw[n, j] = sum_k x[n,k] * W1[j,k]   (x:[N,128], W1:[128,128])
// One wave computes one 16x16 tile using V_WMMA_F32_16X16X4_F32.
// A (16x4 f32) lane layout: lanes 0-15 -> M=lane, VGPR0/1 = K {k,k+1};
//                           lanes 16-31 -> M=lane-16, VGPR0/1 = K {k+2,k+3}.
// B (4x16 f32) lane layout: lanes 0-15 -> N=lane, rows {k,k+1};
//                           lanes 16-31 -> N=lane-16, rows {k+2,k+3}.
// Since B[k][n] = W1[n][k], both A and B are contiguous float2 loads.
// ---------------------------------------------------------------------------
__global__ void gemm1_wmma(const float* __restrict__ x,
                           const float* __restrict__ W1,
                           float* __restrict__ xw, int N) {
    const int lane = threadIdx.x & 31;
    const int wave = threadIdx.x >> 5;     // 0..7 -> column tile (128/16 = 8)
    const int row0 = blockIdx.x * 16;
    const int col0 = wave * 16;
    const int half = lane >> 4;            // 0 or 1
    const int idx  = lane & 15;

    int arow = row0 + idx;
    if (arow >= N) arow = N - 1;           // clamp (N is a multiple of 16 here)
    const float* aptr = x  + (size_t)arow        * 128 + 2 * half;
    const float* bptr = W1 + (size_t)(col0 + idx) * 128 + 2 * half;

    v8f acc = {};
    #pragma unroll
    for (int k = 0; k < 128; k += 4) {
        v2f a = *(const v2f*)(aptr + k);
        v2f b = *(const v2f*)(bptr + k);
        acc = __builtin_amdgcn_wmma_f32_16x16x4_f32(
            false, a, false, b, (short)0, acc, false, false);
    }

    // C/D layout: VGPR r, lanes 0-15 -> M=r, lanes 16-31 -> M=r+8; N=idx
    #pragma unroll
    for (int r = 0; r < 8; ++r) {
        int row = row0 + r + half * 8;
        if (row < N) xw[(size_t)row * 128 + col0 + idx] = acc[r];
    }
}

// ---------------------------------------------------------------------------
// Per-node attention scores: a_src[n,h] = sum_c xw[n,h,c]*att_src[h,c]
// ---------------------------------------------------------------------------
__global__ void attn_scores1(const float* __restrict__ xw,
                             const float* __restrict__ att_src,
                             const float* __restrict__ att_dst,
                             float* __restrict__ asrc,
                             float* __restrict__ adst, int N) {
    int t = blockIdx.x * blockDim.x + threadIdx.x;   // t = n*4 + h
    if (t >= N * 4) return;
    int n = t >> 2, h = t & 3;
    const float* v  = xw + (size_t)n * 128 + h * 32;
    const float* as = att_src + h * 32;
    const float* ad = att_dst + h * 32;
    float s = 0.f, d = 0.f;
    #pragma unroll
    for (int c = 0; c < 32; ++c) {
        float xv = v[c];
        s += xv * as[c];
        d += xv * ad[c];
    }
    asrc[t] = s;
    adst[t] = d;
}

// ---------------------------------------------------------------------------
// Float atomic max via sign-split int/uint trick (init buffer to -inf)
// ---------------------------------------------------------------------------
__device__ __forceinline__ void atomicMaxF(float* addr, float v) {
    if (v >= 0.0f)
        atomicMax((int*)addr, __float_as_int(v));
    else
        atomicMin((unsigned int*)addr, __float_as_uint(v));
}

__device__ __forceinline__ void edge_ids(const int* __restrict__ ei, int e,
                                         int E, int& s, int& d) {
    if (e < E) { s = ei[e]; d = ei[E + e]; }
    else       { s = d = e - E; }          // self loops appended
}

__device__ __forceinline__ float lrelu(float a) {
    return a > 0.f ? a : NEG_SLOPE * a;
}

// Layer-1 edge pass 1: segment max of leaky-relu attention logits
__global__ void edge_max1(const int* __restrict__ ei,
                          const float* __restrict__ asrc,
                          const float* __restrict__ adst,
                          float* __restrict__ amax, int E, int N) {
    int e = blockIdx.x * blockDim.x + threadIdx.x;
    if (e >= E + N) return;
    int s, d; edge_ids(ei, e, E, s, d);
    #pragma unroll
    for (int h = 0; h < 4; ++h) {
        float a = lrelu(asrc[s * 4 + h] + adst[d * 4 + h]);
        atomicMaxF(&amax[d * 4 + h], a);
    }
}

// Layer-1 edge pass 2: denom[dst,h] += exp(alpha - amax[dst,h])
__global__ void edge_den1(const int* __restrict__ ei,
                          const float* __restrict__ asrc,
                          const float* __restrict__ adst,
                          const float* __restrict__ amax,
                          float* __restrict__ denom, int E, int N) {
    int e = blockIdx.x * blockDim.x + threadIdx.x;
    if (e >= E + N) return;
    int s, d; edge_ids(ei, e, E, s, d);
    #pragma unroll
    for (int h = 0; h < 4; ++h) {
        float a = lrelu(asrc[s * 4 + h] + adst[d * 4 + h]);
        atomicAdd(&denom[d * 4 + h], expf(a - amax[d * 4 + h]));
    }
}

// Layer-1 edge pass 3: out[dst, j] += xw[src, j] * coef(e, j>>5)
// 128 threads per edge (one per output feature); coef loads hit L2.
__global__ void edge_msg1(const int* __restrict__ ei,
                          const float* __restrict__ asrc,
                          const float* __restrict__ adst,
                          const float* __restrict__ amax,
                          const float* __restrict__ denom,
                          const float* __restrict__ xw,
                          float* __restrict__ out, int E, int N) {
    long long t = (long long)blockIdx.x * blockDim.x + threadIdx.x;
    long long total = (long long)(E + N) * 128;
    if (t >= total) return;
    int e = (int)(t >> 7);
    int j = (int)(t & 127);
    int h = j >> 5;
    int s, d; edge_ids(ei, e, E, s, d);
    float a    = lrelu(asrc[s * 4 + h] + adst[d * 4 + h]);
    float coef = expf(a - amax[d * 4 + h]) / (denom[d * 4 + h] + 1e-16f);
    atomicAdd(&out[(size_t)d * 128 + j], xw[(size_t)s * 128 + j] * coef);
}

// bias + relu in place: out1 -> h features
__global__ void bias_relu(float* __restrict__ p,
                          const float* __restrict__ b, int N) {
    long long t = (long long)blockIdx.x * blockDim.x + threadIdx.x;
    if (t >= (long long)N * 128) return;
    float v = p[t] + b[t & 127];
    p[t] = v > 0.f ? v : 0.f;
}

// ---------------------------------------------------------------------------
// Layer-2 GEMV: xw2[n] = dot(h[n,:], W2) ; one wave per node
// ---------------------------------------------------------------------------
__global__ void gemv2(const float* __restrict__ h,
                      const float* __restrict__ W2,
                      const float* __restrict__ as2p,
                      const float* __restrict__ ad2p,
                      float* __restrict__ xw2,
                      float* __restrict__ asrc2,
                      float* __restrict__ adst2, int N) {
    int gtid = blockIdx.x * blockDim.x + threadIdx.x;
    int node = gtid >> 5;
    int lane = threadIdx.x & 31;
    if (node >= N) return;
    const float4 a = ((const float4*)(h + (size_t)node * 128))[lane];
    const float4 b = ((const float4*)W2)[lane];
    float s = a.x * b.x + a.y * b.y + a.z * b.z + a.w * b.w;
    #pragma unroll
    for (int off = 16; off > 0; off >>= 1) s += __shfl_xor(s, off, 32);
    if (lane == 0) {
        xw2[node]   = s;
        asrc2[node] = s * as2p[0];
        adst2[node] = s * ad2p[0];
    }
}

// Layer-2 edge passes (single head, scalar features)
__global__ void edge_max2(const int* __restrict__ ei,
                          const float* __restrict__ asrc,
                          const float* __restrict__ adst,
                          float* __restrict__ amax, int E, int N) {
    int e = blockIdx.x * blockDim.x + threadIdx.x;
    if (e >= E + N) return;
    int s, d; edge_ids(ei, e, E, s, d);
    atomicMaxF(&amax[d], lrelu(asrc[s] + adst[d]));
}

__global__ void edge_den2(const int* __restrict__ ei,
                          const float* __restrict__ asrc,
                          const float* __restrict__ adst,
                          const float* __restrict__ amax,
                          float* __restrict__ denom, int E, int N) {
    int e = blockIdx.x * blockDim.x + threadIdx.x;
    if (e >= E + N) return;
    int s, d; edge_ids(ei, e, E, s, d);
    atomicAdd(&denom[d], expf(lrelu(asrc[s] + adst[d]) - amax[d]));
}

__global__ void edge_msg2(const int* __restrict__ ei,
                          const float* __restrict__ asrc,
                          const float* __restrict__ adst,
                          const float* __restrict__ amax,
                          const float* __restrict__ denom,
                          const float* __restrict__ xw2,
                          float* __restrict__ out, int E, int N) {
    int e = blockIdx.x * blockDim.x + threadIdx.x;
    if (e >= E + N) return;
    int s, d; edge_ids(ei, e, E, s, d);
    float coef = expf(lrelu(asrc[s] + adst[d]) - amax[d]) / (denom[d] + 1e-16f);
    atomicAdd(&out[d], xw2[s] * coef);
}

__global__ void add_bias_out(float* __restrict__ out,
                             const float* __restrict__ b2, int N) {
    int t = blockIdx.x * blockDim.x + threadIdx.x;
    if (t < N) out[t] += b2[0];
}

// ---------------------------------------------------------------------------
// launch
// ---------------------------------------------------------------------------
static inline size_t align256(size_t x) { return (x + 255) & ~(size_t)255; }

extern "C" void kernel_launch(void* const* d_in, const int* in_sizes, int n_in,
                              void* d_out, int out_size, void* d_ws, size_t ws_size,
                              hipStream_t stream) {
    const int N = in_sizes[0] / 128;
    const int E = in_sizes[1] / 2;

    const float* x        = (const float*)d_in[0];
    const int*   ei       = (const int*)  d_in[1];
    const float* W1       = (const float*)d_in[2];
    const float* att_src1 = (const float*)d_in[3];
    const float* att_dst1 = (const float*)d_in[4];
    const float* b1       = (const float*)d_in[5];
    const float* W2       = (const float*)d_in[6];
    const float* as2      = (const float*)d_in[7];
    const float* ad2      = (const float*)d_in[8];
    const float* b2       = (const float*)d_in[9];
    float*       out      = (float*)d_out;

    // workspace carve-out
    char* ws = (char*)d_ws;
    size_t off = 0;
    auto take = [&](size_t elems) {
        float* p = (float*)(ws + off);
        off += align256(elems * sizeof(float));
        return p;
    };
    float* xw1    = take((size_t)N * 128);
    float* out1   = take((size_t)N * 128);   // layer-1 accum -> relu'd features
    float* asrc1  = take((size_t)N * 4);
    float* adst1  = take((size_t)N * 4);
    float* amax1  = take((size_t)N * 4);
    float* den1   = take((size_t)N * 4);
    float* xw2    = take((size_t)N);
    float* asrc2  = take((size_t)N);
    float* adst2  = take((size_t)N);
    float* amax2  = take((size_t)N);
    float* den2   = take((size_t)N);
    (void)ws_size; (void)n_in; (void)out_size;

    const int B = 256;
    auto blocks = [&](long long n) { return (unsigned)((n + B - 1) / B); };
    const long long EN = (long long)E + N;

    // init accumulators (every call: graph replays must be self-consistent)
    fill_f32<<<blocks((long long)N * 128), B, 0, stream>>>(out1, (long long)N * 128, 0.f);
    fill_f32<<<blocks(N * 4), B, 0, stream>>>(amax1, N * 4, -INFINITY);
    fill_f32<<<blocks(N * 4), B, 0, stream>>>(den1,  N * 4, 0.f);
    fill_f32<<<blocks(N), B, 0, stream>>>(amax2, N, -INFINITY);
    fill_f32<<<blocks(N), B, 0, stream>>>(den2,  N, 0.f);
    fill_f32<<<blocks(N), B, 0, stream>>>(out,   N, 0.f);

    // layer 1
    gemm1_wmma<<<(N + 15) / 16, 256, 0, stream>>>(x, W1, xw1, N);
    attn_scores1<<<blocks(N * 4), B, 0, stream>>>(xw1, att_src1, att_dst1, asrc1, adst1, N);
    edge_max1<<<blocks(EN), B, 0, stream>>>(ei, asrc1, adst1, amax1, E, N);
    edge_den1<<<blocks(EN), B, 0, stream>>>(ei, asrc1, adst1, amax1, den1, E, N);
    edge_msg1<<<blocks(EN * 128), B, 0, stream>>>(ei, asrc1, adst1, amax1, den1, xw1, out1, E, N);
    bias_relu<<<blocks((long long)N * 128), B, 0, stream>>>(out1, b1, N);

    // layer 2
    gemv2<<<blocks((long long)N * 32), B, 0, stream>>>(out1, W2, as2, ad2, xw2, asrc2, adst2, N);
    edge_max2<<<blocks(EN), B, 0, stream>>>(ei, asrc2, adst2, amax2, E, N);
    edge_den2<<<blocks(EN), B, 0, stream>>>(ei, asrc2, adst2, amax2, den2, E, N);
    edge_msg2<<<blocks(EN), B, 0, stream>>>(ei, asrc2, adst2, amax2, den2, xw2, out, E, N);
    add_bias_out<<<blocks(N), B, 0, stream>>>(out, b2, N);
}